// MinimalTransformerWithKVCache_13176959664585
// MI455X (gfx1250) — compile-verified
//
#include <hip/hip_runtime.h>
#include <hip/hip_bf16.h>

typedef __attribute__((ext_vector_type(2))) float v2f;
typedef __attribute__((ext_vector_type(4))) float v4f;
typedef __attribute__((ext_vector_type(8))) float v8f;

#define T_L   4
#define T_B   8
#define T_H   16
#define T_S   4096
#define T_HD  64
#define T_Q   4
#define T_D   1024
#define T_V   32000
#define CS    1024   // key chunk per attention block
#define TS    128    // key sub-tile staged in LDS
#define KPAD  68     // padded LDS row (272B = 17*16B -> B128-aligned rows)
#define NCH   5      // 4 past chunks + 1 new-token tail chunk
#define ATT_SCALE 0.125f

// LDS byte offset of a pointer into a __shared__ array: flat->local
// addrspacecast truncates to the low 32 bits, which is the LDS offset.
__device__ __forceinline__ unsigned int lds_off(const void* p) {
  return (unsigned int)(unsigned long long)
         (__attribute__((address_space(3))) const char*)p;
}

// -------------------------------------------------------------------------
// Embedding: x[r,:] = emb[id] + pos_emb[pos], r in [0,32)
// -------------------------------------------------------------------------
__global__ __launch_bounds__(256) void embed_kernel(
    const int* __restrict__ ids, const int* __restrict__ pos,
    const float* __restrict__ emb, const float* __restrict__ pemb,
    float* __restrict__ x)
{
  int r  = blockIdx.x;
  int id = ids[r];
  int p  = pos[r];
  int c4 = threadIdx.x << 2;                 // 256 threads * 4 = 1024 = D
  v4f e  = *(const v4f*)(emb  + (size_t)id * T_D + c4);
  v4f pe = *(const v4f*)(pemb + (size_t)p  * T_D + c4);
  *(v4f*)(x + (size_t)r * T_D + c4) = e + pe;
}

// -------------------------------------------------------------------------
// Skinny GEMM: Y[M=32,N] = X[32,K] * W[N,K]^T (+ residual R, optional relu)
// One wave per 16x16 output tile, V_WMMA_F32_16X16X4_F32 (exact f32).
// -------------------------------------------------------------------------
__global__ __launch_bounds__(256) void gemm_xwt(
    const float* __restrict__ X, const float* __restrict__ W,
    float* Y, const float* R, int N, int K, int relu)
{
  const int M = 32;
  int lane   = threadIdx.x & 31;
  int wid    = blockIdx.x * (blockDim.x >> 5) + (threadIdx.x >> 5);
  int ntiles = N >> 4;
  int total  = ntiles * (M >> 4);
  if (wid >= total) return;                  // whole-wave exit: EXEC stays all-1s
  int tn   = wid % ntiles;
  int tm   = wid / ntiles;
  int half = lane >> 4;                      // 0 or 1 -> k-slot pair
  int l16  = lane & 15;

  const float* xr = X + (size_t)(tm * 16 + l16) * K + half * 4;
  const float* wr = W + (size_t)(tn * 16 + l16) * K + half * 4;

  v8f acc = {};
  #pragma unroll 2
  for (int k = 0; k < K; k += 8) {
    v4f a = *(const v4f*)(xr + k);           // global_load_b128
    v4f b = *(const v4f*)(wr + k);
    v2f a0 = {a.x, a.y}, a1 = {a.z, a.w};
    v2f b0 = {b.x, b.y}, b1 = {b.z, b.w};
    acc = __builtin_amdgcn_wmma_f32_16x16x4_f32(false, a0, false, b0,
                                                (short)0, acc, false, false);
    acc = __builtin_amdgcn_wmma_f32_16x16x4_f32(false, a1, false, b1,
                                                (short)0, acc, false, false);
  }

  // C/D layout: VGPR v -> row tm*16 + half*8 + v, col tn*16 + l16
  int cn = tn * 16 + l16;
  int m0 = tm * 16 + half * 8;
  #pragma unroll
  for (int v = 0; v < 8; ++v) {
    float val = acc[v];
    size_t off = (size_t)(m0 + v) * N + cn;
    if (R)    val += R[off];
    if (relu) val  = fmaxf(val, 0.0f);
    Y[off] = val;
  }
}

// -------------------------------------------------------------------------
// Flash-decoding attention partial: one block per (b,h,chunk).
// K/V tiles staged via GLOBAL_LOAD_ASYNC_TO_LDS_B128 (no VGPR roundtrip),
// present-KV copy fused via GLOBAL_STORE_ASYNC_FROM_LDS_B128 running in
// the background of the softmax math. ASYNCcnt-tracked, wave32.
// -------------------------------------------------------------------------
__global__ __launch_bounds__(256) void attn_partial(
    const float* __restrict__ qkv,   // [B,Q,3D]
    const float* __restrict__ pk,    // layer past keys [B,H,S,HD]
    const float* __restrict__ pv,
    float* __restrict__ prk,         // layer present keys [B,H,S+Q,HD]
    float* __restrict__ prv,
    float* __restrict__ opart,       // [B*H, NCH, Q, HD]
    float* __restrict__ mpart,       // [B*H, NCH, Q]
    float* __restrict__ lpart)
{
  __shared__ float q_s[T_Q][T_HD];
  __shared__ float k_s[TS][KPAD];    // 16B-aligned rows for async B128
  __shared__ float v_s[TS][KPAD];
  __shared__ float p_s[T_Q][TS];
  __shared__ float m_s[T_Q], l_s[T_Q], f_s[T_Q];

  int tid = threadIdx.x;
  int blk = blockIdx.x;
  int ch  = blk % NCH;
  int bh  = blk / NCH;
  int h   = bh % T_H;
  int b   = bh / T_H;
  int warp = tid >> 5, lane = tid & 31;
  int myq  = tid >> 6, myd  = tid & 63;

  const float* pkb  = pk  + (size_t)bh * T_S * T_HD;
  const float* pvb  = pv  + (size_t)bh * T_S * T_HD;
  float*       prkb = prk + (size_t)bh * (T_S + T_Q) * T_HD;
  float*       prvb = prv + (size_t)bh * (T_S + T_Q) * T_HD;
  const float* qkvb = qkv + (size_t)b * T_Q * 3 * T_D;

  { // load q vectors for this head
    int qq = tid >> 6, d = tid & 63;
    q_s[qq][d] = qkvb[(size_t)qq * 3 * T_D + h * T_HD + d];
  }
  if (tid < T_Q) { m_s[tid] = -3.0e38f; l_s[tid] = 0.0f; }
  float o_acc = 0.0f;
  __syncthreads();

  int kbeg = ch * CS;
  int kend = min(T_S + T_Q, kbeg + CS);

  for (int t0 = kbeg; t0 < kend; t0 += TS) {
    int ts = min(TS, kend - t0);

    // ---- async-stage K/V tile: global -> LDS, 16B per lane per op ----
    for (int idx = tid; idx < ts * 16; idx += 256) {
      int row = idx >> 4;
      int c4  = (idx & 15) << 2;
      int j   = t0 + row;
      const float *gk, *gv;
      if (j < T_S) {
        gk = pkb + (size_t)j * T_HD + c4;
        gv = pvb + (size_t)j * T_HD + c4;
      } else {
        int qq = j - T_S;                      // new token from qkv buffer
        const float* base = qkvb + (size_t)qq * 3 * T_D + h * T_HD + c4;
        gk = base + T_D;
        gv = base + 2 * T_D;
      }
      unsigned int lk = lds_off(&k_s[row][c4]);
      unsigned int lv = lds_off(&v_s[row][c4]);
      asm volatile("global_load_async_to_lds_b128 %0, %1, off"
                   :: "v"(lk), "v"(gk) : "memory");
      asm volatile("global_load_async_to_lds_b128 %0, %1, off"
                   :: "v"(lv), "v"(gv) : "memory");
    }
    asm volatile("s_wait_asynccnt 0x0" ::: "memory");
    __syncthreads();

    // ---- fused present-KV copy: LDS -> global, runs under the math ----
    for (int idx = tid; idx < ts * 16; idx += 256) {
      int row = idx >> 4;
      int c4  = (idx & 15) << 2;
      int j   = t0 + row;
      float* dk = prkb + (size_t)j * T_HD + c4;
      float* dv = prvb + (size_t)j * T_HD + c4;
      unsigned int lk = lds_off(&k_s[row][c4]);
      unsigned int lv = lds_off(&v_s[row][c4]);
      asm volatile("global_store_async_from_lds_b128 %0, %1, off"
                   :: "v"(dk), "v"(lk) : "memory");
      asm volatile("global_store_async_from_lds_b128 %0, %1, off"
                   :: "v"(dv), "v"(lv) : "memory");
    }

    // ---- scores: thread t -> key row t, 4 dots of 64 ----
    if (tid < ts) {
      float s0 = 0.f, s1 = 0.f, s2 = 0.f, s3 = 0.f;
      #pragma unroll 8
      for (int dd = 0; dd < T_HD; ++dd) {
        float kvv = k_s[tid][dd];
        s0 = fmaf(q_s[0][dd], kvv, s0);
        s1 = fmaf(q_s[1][dd], kvv, s1);
        s2 = fmaf(q_s[2][dd], kvv, s2);
        s3 = fmaf(q_s[3][dd], kvv, s3);
      }
      p_s[0][tid] = s0 * ATT_SCALE;  p_s[1][tid] = s1 * ATT_SCALE;
      p_s[2][tid] = s2 * ATT_SCALE;  p_s[3][tid] = s3 * ATT_SCALE;
    }
    __syncthreads();

    // ---- tile max per q (warp w handles query w) ----
    if (warp < T_Q) {
      float mx = -3.0e38f;
      for (int j = lane; j < ts; j += 32) mx = fmaxf(mx, p_s[warp][j]);
      for (int off = 16; off > 0; off >>= 1)
        mx = fmaxf(mx, __shfl_xor(mx, off, 32));
      if (lane == 0) {
        float mo = m_s[warp];
        float mn = fmaxf(mo, mx);
        f_s[warp] = __expf(mo - mn);
        m_s[warp] = mn;
      }
    }
    __syncthreads();

    // ---- exponentiate in place ----
    if (tid < ts) {
      #pragma unroll
      for (int q = 0; q < T_Q; ++q)
        p_s[q][tid] = __expf(p_s[q][tid] - m_s[q]);
    }
    __syncthreads();

    // ---- l update ----
    if (warp < T_Q) {
      float sm = 0.f;
      for (int j = lane; j < ts; j += 32) sm += p_s[warp][j];
      for (int off = 16; off > 0; off >>= 1) sm += __shfl_xor(sm, off, 32);
      if (lane == 0) l_s[warp] = l_s[warp] * f_s[warp] + sm;
    }
    __syncthreads();

    // ---- o update: thread (q,d) accumulates over tile keys ----
    float f = f_s[myq];
    o_acc *= f;
    #pragma unroll 4
    for (int j = 0; j < ts; ++j)
      o_acc = fmaf(p_s[myq][j], v_s[j][myd], o_acc);

    // present-copy stores must finish before next tile reuses the LDS
    asm volatile("s_wait_asynccnt 0x0" ::: "memory");
    __syncthreads();
  }

  size_t pbase = ((size_t)bh * NCH + ch) * T_Q;
  opart[(pbase + myq) * T_HD + myd] = o_acc;
  if (tid < T_Q) { mpart[pbase + tid] = m_s[tid]; lpart[pbase + tid] = l_s[tid]; }
}

// -------------------------------------------------------------------------
// Combine NCH partials per (b,h); write o transposed to [B,Q,D] layout.
// -------------------------------------------------------------------------
__global__ __launch_bounds__(256) void attn_reduce(
    const float* __restrict__ opart, const float* __restrict__ mpart,
    const float* __restrict__ lpart, float* __restrict__ obuf)
{
  int bh = blockIdx.x;
  int h  = bh % T_H, b = bh / T_H;
  int tid = threadIdx.x;
  int q = tid >> 6, d = tid & 63;
  size_t base = (size_t)bh * NCH;

  float M = -3.0e38f;
  #pragma unroll
  for (int c = 0; c < NCH; ++c)
    M = fmaxf(M, mpart[(base + c) * T_Q + q]);
  float L = 0.f, o = 0.f;
  #pragma unroll
  for (int c = 0; c < NCH; ++c) {
    float w = __expf(mpart[(base + c) * T_Q + q] - M);
    L += w * lpart[(base + c) * T_Q + q];
    o += w * opart[((base + c) * T_Q + q) * T_HD + d];
  }
  obuf[(size_t)(b * T_Q + q) * T_D + h * T_HD + d] = o / L;
}

// -------------------------------------------------------------------------
extern "C" void kernel_launch(void* const* d_in, const int* in_sizes, int n_in,
                              void* d_out, int out_size, void* d_ws, size_t ws_size,
                              hipStream_t stream)
{
  const int*   input_ids    = (const int*)d_in[0];
  const int*   position_ids = (const int*)d_in[1];
  const float* past_k = (const float*)d_in[2];
  const float* past_v = (const float*)d_in[3];
  const float* emb    = (const float*)d_in[4];
  const float* pemb   = (const float*)d_in[5];
  const float* qkv_w  = (const float*)d_in[6];
  const float* out_w  = (const float*)d_in[7];
  const float* fc1_w  = (const float*)d_in[8];
  const float* fc2_w  = (const float*)d_in[9];
  const float* lm_w   = (const float*)d_in[10];

  float* logits = (float*)d_out;                                     // [B,Q,V]
  float* pres_k = logits + (size_t)T_B * T_Q * T_V;                  // [L,B,H,S+Q,HD]
  float* pres_v = pres_k + (size_t)T_L * T_B * T_H * (T_S + T_Q) * T_HD;

  float* ws     = (float*)d_ws;
  float* x      = ws;  ws += 32 * T_D;
  float* qkvbuf = ws;  ws += 32 * 3 * T_D;
  float* hbuf   = ws;  ws += 32 * 4 * T_D;
  float* obuf   = ws;  ws += 32 * T_D;
  float* opart  = ws;  ws += (size_t)T_B * T_H * NCH * T_Q * T_HD;
  float* mpart  = ws;  ws += T_B * T_H * NCH * T_Q;
  float* lpart  = ws;  ws += T_B * T_H * NCH * T_Q;

  embed_kernel<<<32, 256, 0, stream>>>(input_ids, position_ids, emb, pemb, x);

  for (int l = 0; l < T_L; ++l) {
    const float* qw  = qkv_w + (size_t)l * 3 * T_D * T_D;
    const float* ow  = out_w + (size_t)l * T_D * T_D;
    const float* f1  = fc1_w + (size_t)l * 4 * T_D * T_D;
    const float* f2  = fc2_w + (size_t)l * T_D * 4 * T_D;
    const float* pkl = past_k + (size_t)l * T_B * T_H * T_S * T_HD;
    const float* pvl = past_v + (size_t)l * T_B * T_H * T_S * T_HD;
    float* prkl = pres_k + (size_t)l * T_B * T_H * (T_S + T_Q) * T_HD;
    float* prvl = pres_v + (size_t)l * T_B * T_H * (T_S + T_Q) * T_HD;

    // waves = (N/16)*(32/16); blocks = waves/8
    gemm_xwt<<<48, 256, 0, stream>>>(x, qw, qkvbuf, nullptr, 3 * T_D, T_D, 0);
    attn_partial<<<T_B * T_H * NCH, 256, 0, stream>>>(qkvbuf, pkl, pvl, prkl,
                                                      prvl, opart, mpart, lpart);
    attn_reduce<<<T_B * T_H, 256, 0, stream>>>(opart, mpart, lpart, obuf);
    gemm_xwt<<<16, 256, 0, stream>>>(obuf, ow, x, x, T_D, T_D, 0);
    gemm_xwt<<<64, 256, 0, stream>>>(x, f1, hbuf, nullptr, 4 * T_D, T_D, 1);
    gemm_xwt<<<16, 256, 0, stream>>>(hbuf, f2, x, x, T_D, 4 * T_D, 0);
  }
  gemm_xwt<<<500, 256, 0, stream>>>(x, lm_w, logits, nullptr, T_V, T_D, 0);
}